// StateSpaceModule_14912126452378
// MI455X (gfx1250) — compile-verified
//
#include <hip/hip_runtime.h>

typedef __attribute__((ext_vector_type(2))) float v2f;
typedef __attribute__((ext_vector_type(4))) float v4f;
typedef __attribute__((ext_vector_type(8))) float v8f;

#define DIM   128
#define NEXP  16

// workspace layout (ints)
#define WS_COUNTS 0    // 16 counts
#define WS_OFFS   16   // 17 padded prefix offsets
#define WS_NTILES 33   // 1 tile count
#define WS_FILL   34   // 16 fill cursors
#define WS_ROWS   64   // row-index buckets, padded with -1

__global__ void ssm_init(int* ws, int rowsCap) {
  int i = blockIdx.x * blockDim.x + threadIdx.x;
  if (i < 64) ws[i] = 0;
  if (i < rowsCap) ws[WS_ROWS + i] = -1;
}

__global__ void ssm_count(const int* __restrict__ sidx, int* ws, int n) {
  int i = blockIdx.x * blockDim.x + threadIdx.x;
  if (i < n) atomicAdd(&ws[WS_COUNTS + sidx[i]], 1);
}

__global__ void ssm_offsets(int* ws) {
  if (threadIdx.x == 0 && blockIdx.x == 0) {
    int off = 0;
    for (int e = 0; e < NEXP; ++e) {
      ws[WS_OFFS + e] = off;
      off += ((ws[WS_COUNTS + e] + 15) >> 4) << 4;   // pad bucket to 16 rows
    }
    ws[WS_OFFS + NEXP] = off;
    ws[WS_NTILES] = off >> 4;
  }
}

__global__ void ssm_scatter(const int* __restrict__ sidx, int* ws, int n) {
  int i = blockIdx.x * blockDim.x + threadIdx.x;
  if (i < n) {
    int e = sidx[i];
    int pos = atomicAdd(&ws[WS_FILL + e], 1);
    ws[WS_ROWS + ws[WS_OFFS + e] + pos] = i;
  }
}

// One block = one 16-row tile of one expert. 4 waves; wave w owns N-tiles {2w,2w+1}.
__launch_bounds__(128)
__global__ void ssm_gemm(const float* __restrict__ x,
                         const int*   __restrict__ sidx,
                         const float* __restrict__ A_all,
                         const float* __restrict__ C_all,
                         const int*   __restrict__ ws,
                         float*       __restrict__ out)
{
  __shared__ float xs[16 * 132];   // stride 132 -> conflict-free b64 fragment reads
  __shared__ float ss[16 * 132];

  const int t = blockIdx.x;
  if (t >= ws[WS_NTILES]) return;               // wave-uniform exit

  const int* rows = ws + WS_ROWS;
  const int base  = t << 4;
  const int e     = sidx[rows[base]];           // first row of a tile is always valid
  const float* __restrict__ Ae = A_all + (size_t)e * DIM * DIM;
  const float* __restrict__ Ce = C_all + (size_t)e * DIM * DIM;

  const int tid = threadIdx.x;
  const int w   = tid >> 5;        // wave id 0..3
  const int ln  = tid & 31;
  const int hh  = ln >> 4;         // lane half: selects K pair / M half
  const int lm  = ln & 15;

  // ---- stage gathered X tile [16 x 128] into LDS ----
  {
    const int ri  = tid >> 3;              // 0..15 (row in tile)
    const int ci  = (tid & 7) << 4;        // 0,16,...,112
    const int row = rows[base + ri];
    float* dst = &xs[ri * 132 + ci];
    if (row >= 0) {
      const v4f* src = (const v4f*)(x + (size_t)row * DIM + ci);
#pragma unroll
      for (int j = 0; j < 4; ++j) *(v4f*)(dst + 4 * j) = src[j];
    } else {
#pragma unroll
      for (int j = 0; j < 16; ++j) dst[j] = 0.0f;   // padded row
    }
  }
  __syncthreads();

  const int nt0 = w << 1;
  const int n0  = (nt0 << 4) + lm;
  const int n1  = n0 + 16;

  // ---- GEMM1: S = X @ Ae (fp32 WMMA, K=128 in 32 steps of 4) ----
  v8f acc0 = {}; v8f acc1 = {};
  {
    const float* ap  = &xs[lm * 132 + (hh << 1)];  // A-operand: lane=(k/2)*16+m
    const float* b0p = Ae + n0;                    // B-operand: lane=(k/2)*16+n
    const float* b1p = Ae + n1;
#pragma unroll 4
    for (int kk = 0; kk < 32; ++kk) {
      const int k0 = kk << 2;
      v2f a = *(const v2f*)(ap + k0);
      const int kr = (k0 + (hh << 1)) * DIM;
      v2f b0, b1;
      b0.x = b0p[kr]; b0.y = b0p[kr + DIM];
      b1.x = b1p[kr]; b1.y = b1p[kr + DIM];
      acc0 = __builtin_amdgcn_wmma_f32_16x16x4_f32(false, a, false, b0,
                                                   (short)0, acc0, false, false);
      acc1 = __builtin_amdgcn_wmma_f32_16x16x4_f32(false, a, false, b1,
                                                   (short)0, acc1, false, false);
    }
  }

  // ---- ReLU + stash S in LDS (D-layout -> row-major, conflict-free) ----
#pragma unroll
  for (int r = 0; r < 8; ++r) {
    const int m = r + (hh << 3);
    ss[m * 132 + n0] = fmaxf(acc0[r], 0.0f);
    ss[m * 132 + n1] = fmaxf(acc1[r], 0.0f);
  }
  __syncthreads();

  // ---- GEMM2: O = S @ Ce ----
  v8f o0 = {}; v8f o1 = {};
  {
    const float* ap  = &ss[lm * 132 + (hh << 1)];
    const float* b0p = Ce + n0;
    const float* b1p = Ce + n1;
#pragma unroll 4
    for (int kk = 0; kk < 32; ++kk) {
      const int k0 = kk << 2;
      v2f a = *(const v2f*)(ap + k0);
      const int kr = (k0 + (hh << 1)) * DIM;
      v2f b0, b1;
      b0.x = b0p[kr]; b0.y = b0p[kr + DIM];
      b1.x = b1p[kr]; b1.y = b1p[kr + DIM];
      o0 = __builtin_amdgcn_wmma_f32_16x16x4_f32(false, a, false, b0,
                                                 (short)0, o0, false, false);
      o1 = __builtin_amdgcn_wmma_f32_16x16x4_f32(false, a, false, b1,
                                                 (short)0, o1, false, false);
    }
  }

  // ---- masked scatter of output rows ----
#pragma unroll
  for (int r = 0; r < 8; ++r) {
    const int m   = r + (hh << 3);
    const int row = rows[base + m];
    if (row >= 0) {
      out[(size_t)row * DIM + n0] = o0[r];
      out[(size_t)row * DIM + n1] = o1[r];
    }
  }
}

extern "C" void kernel_launch(void* const* d_in, const int* in_sizes, int n_in,
                              void* d_out, int out_size, void* d_ws, size_t ws_size,
                              hipStream_t stream) {
  const float* x     = (const float*)d_in[0];
  const int*   sidx  = (const int*)d_in[1];
  const float* A_all = (const float*)d_in[2];
  const float* C_all = (const float*)d_in[3];
  float* out = (float*)d_out;
  int*   ws  = (int*)d_ws;

  const int batch    = in_sizes[1];
  const int maxTiles = (batch + 15) / 16 + NEXP;   // worst-case padding
  const int rowsCap  = maxTiles * 16;

  ssm_init   <<<(rowsCap + 255) / 256, 256, 0, stream>>>(ws, rowsCap);
  ssm_count  <<<(batch + 255) / 256, 256, 0, stream>>>(sidx, ws, batch);
  ssm_offsets<<<1, 32, 0, stream>>>(ws);
  ssm_scatter<<<(batch + 255) / 256, 256, 0, stream>>>(sidx, ws, batch);
  ssm_gemm   <<<maxTiles, 128, 0, stream>>>(x, sidx, A_all, C_all, ws, out);
}